// PoissonIntervalEncoder_74637941670502
// MI455X (gfx1250) — compile-verified
//
#include <hip/hip_runtime.h>
#include <stdint.h>

// Poisson-interval spike encoder for MI455X (gfx1250).
// Store-bandwidth-bound: 262 MB output / 23.3 TB/s ~ 11 us floor.
// CDNA5 paths used: TDM tensor_load_to_lds (async tensor DMA -> LDS),
// s_wait_tensorcnt, LDS reads, non-temporal streaming stores.

#define ENC_STEPS 250

typedef __attribute__((ext_vector_type(4))) unsigned int v4u_t;
typedef __attribute__((ext_vector_type(8))) int          v8i_t;
typedef __attribute__((ext_vector_type(4))) int          v4i_t;

__device__ __forceinline__ uint64_t splitmix64(uint64_t x) {
  x += 0x9E3779B97F4A7C15ULL;
  x = (x ^ (x >> 30)) * 0xBF58476D1CE4E5B9ULL;
  x = (x ^ (x >> 27)) * 0x94D049BB133111EBULL;
  return x ^ (x >> 31);
}

// Deterministic per-(unit, counter) standard normal (Box-Muller).
__device__ __forceinline__ float gauss01(uint32_t unit, uint32_t ctr) {
  uint64_t h = splitmix64(((uint64_t)unit << 32) | (uint64_t)ctr);
  uint32_t a = (uint32_t)h;
  uint32_t b = (uint32_t)(h >> 32);
  float u1 = ((float)((a >> 8) + 1u)) * (1.0f / 16777216.0f); // (0,1]
  float u2 = ((float)(b >> 8))        * (1.0f / 16777216.0f); // [0,1)
  float r = sqrtf(fmaxf(-2.0f * __logf(u1), 0.0f));
  return r * __cosf(6.28318530718f * u2);
}

__global__ void __launch_bounds__(256)
poisson_interval_encode(const float* __restrict__ in,
                        const float* __restrict__ fscale,
                        float* __restrict__ out, int n)
{
  __shared__ float tile[256];
  const int tid  = threadIdx.x;
  const int base = blockIdx.x * 256;
  const int unit = base + tid;

  // --- Stage this block's 256-float input tile into LDS via TDM ---
  if (tid == 0) {
    uint64_t gaddr  = (uint64_t)(uintptr_t)(in + base);
    uint32_t ldsoff = (uint32_t)(uintptr_t)(&tile[0]); // low 32 bits = LDS offset
    int rem = n - base; if (rem > 256) rem = 256; if (rem < 0) rem = 0;

    // D# group 0: count=1 (valid), lds_addr, global_addr[56:0], type=2 ("image")
    v4u_t g0;
    g0.x = 1u;
    g0.y = ldsoff;
    g0.z = (uint32_t)gaddr;
    g0.w = (uint32_t)((gaddr >> 32) & 0x01FFFFFFu) | (2u << 30);

    // D# group 1: wg_mask=0, data_size=2 (4B), tensor_dim0=rem (OOB -> zero),
    // tensor_dim1=1, tile_dim0=256, tile_dim1=1, stride0=256.
    v8i_t g1;
    g1[0] = (int)(2u << 16);
    g1[1] = (int)(((uint32_t)rem & 0xFFFFu) << 16);
    g1[2] = (int)(1u << 16);
    g1[3] = (int)(256u << 16);
    g1[4] = 1;
    g1[5] = 256;
    g1[6] = 0;
    g1[7] = 0;

    v4i_t gz4 = {0, 0, 0, 0};
    v8i_t gz8 = {0, 0, 0, 0, 0, 0, 0, 0};
    // clang-23 / therock-10.0 form: 6 args (extra int32x8 group before cpol).
    __builtin_amdgcn_tensor_load_to_lds(g0, g1, gz4, gz4, gz8, 0);
    __builtin_amdgcn_s_wait_tensorcnt(0);
  }
  __syncthreads();

  if (unit >= n) return;

  const float fs   = fscale[0];
  const float x    = tile[tid];       // ds_load from the TDM-staged tile
  const float freq = fs * x;          // Hz

  // Walk spike times: t = next spike step (1..ENC_STEPS valid).
  int t;
  float lam = 0.0f, sq = 0.0f;
  uint32_t ctr = 0;
  if (freq > 0.0f) {
    lam = 1000.0f / freq;             // mean interval in steps (dt = 1 ms)
    sq  = sqrtf(lam);
    float fk = floorf(lam + sq * gauss01((uint32_t)unit, ctr++) + 0.5f);
    if (fk < 1.0f) fk = 1.0f;         // active units wait >= 1 step
    t = (fk > 300.0f) ? (ENC_STEPS + 1) : (int)fk;
  } else {
    t = ENC_STEPS + 1;                // inactive: no spikes at all
  }

  // Emit 250 timesteps as 8 windows of 32 bits; each window lives in one
  // register, so no private arrays / scratch. Stores are coalesced per t
  // (consecutive lanes -> consecutive units) and non-temporal (write-once
  // 262 MB stream).
  float* p = out + unit;
  const size_t stride = (size_t)n;
  #pragma unroll
  for (int w = 0; w < 8; ++w) {
    uint32_t word = 0;
    const int lim = w * 32 + 32;      // spike index (t-1) must be < lim
    while (t <= ENC_STEPS && (t - 1) < lim) {
      word |= 1u << ((t - 1) & 31);
      float fk = floorf(lam + sq * gauss01((uint32_t)unit, ctr++) + 0.5f);
      if (fk < 1.0f) fk = 1.0f;
      if (fk > 300.0f) { t = ENC_STEPS + 1; }   // surely past horizon
      else             { t += (int)fk; }
    }
    const int cnt = (w == 7) ? (ENC_STEPS - 224) : 32;  // last window: 26
    for (int b = 0; b < cnt; ++b) {
      __builtin_nontemporal_store((float)((word >> b) & 1u), p);
      p += stride;
    }
  }
}

extern "C" void kernel_launch(void* const* d_in, const int* in_sizes, int n_in,
                              void* d_out, int out_size, void* d_ws, size_t ws_size,
                              hipStream_t stream) {
  const float* in  = (const float*)d_in[0];   // (64, 4096) float32
  const float* fs  = (const float*)d_in[1];   // scalar freqscale
  float* out       = (float*)d_out;           // (250, 64, 4096) float32
  const int n      = in_sizes[0];             // 262144 units
  const int blocks = (n + 255) / 256;
  hipLaunchKernelGGL(poisson_interval_encode, dim3(blocks), dim3(256), 0, stream,
                     in, fs, out, n);
}